// VectorQuantizer_69518340653127
// MI455X (gfx1250) — compile-verified
//
#include <hip/hip_runtime.h>

// Problem constants (from reference)
#define D       512
#define KCODES  8192
#define NROWS   16384                 // 32 * 512
#define NUMEL   (NROWS * D)           // 8388608

// Main-kernel tiling
#define MB      32                    // rows per workgroup (2 row-groups of 16)
#define CT      256                   // codes per outer iteration (16 tiles of 16; 4 per wave)
#define DC      64                    // d-chunk staged in LDS per inner step
#define XSTRIDE (D + 4)               // 516 -> A reads: bank = (4*row + off) % 64, conflict free
#define ESTRIDE (DC + 4)              // 68  -> B reads: bank = (4*code + off) % 64, conflict free

typedef float v2f __attribute__((ext_vector_type(2)));
typedef float v8f __attribute__((ext_vector_type(8)));
typedef int   v4i __attribute__((vector_size(16)));   // matches builtin param type

// Generic -> named address-space casts. On AMDGPU a generic LDS address's low
// 32 bits ARE the wave-relative LDS offset (ISA 10.2),
// so the integer-truncation cast to AS3 is an exact addrspacecast.
#define AS1V4(p) ((__attribute__((address_space(1))) v4i*)(uintptr_t)(p))
#define AS3V4(p) ((__attribute__((address_space(3))) v4i*)(uintptr_t)(p))

#if __has_builtin(__builtin_amdgcn_global_load_async_to_lds_b128)
#define HAVE_ASYNC_LDS 1
#else
#define HAVE_ASYNC_LDS 0
#endif

__device__ __forceinline__ void wait_asynccnt0() {
#if __has_builtin(__builtin_amdgcn_s_wait_asynccnt)
  __builtin_amdgcn_s_wait_asynccnt(0);
#else
  asm volatile("s_wait_asynccnt 0" ::: "memory");
#endif
}

// ---------------------------------------------------------------------------
// Kernel 1: e_sq[k] = sum_d emb[k][d]^2.  One wave (32 lanes) per code.
// ---------------------------------------------------------------------------
__global__ void __launch_bounds__(256) vq_esq_kernel(const float* __restrict__ emb,
                                                     float* __restrict__ esq) {
  const int code = blockIdx.x * 8 + (threadIdx.x >> 5);
  const int lane = threadIdx.x & 31;
  const float4* e = (const float4*)(emb + (size_t)code * D);
  float s = 0.f;
#pragma unroll
  for (int i = 0; i < D / 4 / 32; ++i) {            // 4 float4 per lane
    float4 v = e[lane + i * 32];
    s += v.x * v.x + v.y * v.y + v.z * v.z + v.w * v.w;
  }
#pragma unroll
  for (int off = 16; off > 0; off >>= 1) s += __shfl_xor(s, off, 32);
  if (lane == 0) esq[code] = s;
}

// ---------------------------------------------------------------------------
// Kernel 2: fused GEMM (f32 WMMA 16x16x4) + row-argmin of (e_sq - 2*dot).
//   256 threads = 8 waves; wave (r,c): r in {0,1} -> 16-row group,
//   c in {0..3} -> 4 independent 16-code tiles (4 WMMA chains, A reused).
// ---------------------------------------------------------------------------
__global__ void __launch_bounds__(256) vq_argmin_kernel(
    const float* __restrict__ x, const float* __restrict__ emb,
    const float* __restrict__ esq, int* __restrict__ idx_out) {
  extern __shared__ float smem[];
  float* xs   = smem;                        // MB * XSTRIDE
  float* es   = xs + MB * XSTRIDE;           // CT * ESTRIDE
  float* redv = es + CT * ESTRIDE;           // MB * 4
  int*   redi = (int*)(redv + MB * 4);       // MB * 4

  const int tid  = threadIdx.x;
  const int wave = tid >> 5;
  const int lane = tid & 31;
  const int r    = wave >> 2;                // row-group 0..1
  const int c    = wave & 3;                 // wave's group of 4 col tiles
  const int half = lane >> 4;                // lane half (ISA A/B/C layouts)
  const int l15  = lane & 15;
  const int row0 = blockIdx.x * MB;

  // Stage 32 x-rows into LDS once; reused for all 8192 codes.
  for (int i = tid; i < MB * D / 4; i += 256) {
    const int rr = i / (D / 4);
    const int dd = (i % (D / 4)) * 4;
    float4 v = *(const float4*)(x + (size_t)(row0 + rr) * D + dd);
    *(float4*)(xs + rr * XSTRIDE + dd) = v;
  }

  float minval[8];
  int   minidx[8];
#pragma unroll
  for (int v = 0; v < 8; ++v) { minval[v] = 3.402823466e38f; minidx[v] = 0; }

  // ISA 32-bit A 16x4 layout: VGPR0 holds K = 2*half, VGPR1 holds K = 2*half+1.
  // B 4x16 mirrors it (row striped across lanes, halves offset by 2 in K).
  const float* xrow = xs + (16 * r + l15) * XSTRIDE;        // A: this lane's row
  const float* erow = es + (c * 64 + l15) * ESTRIDE;        // B tile 0 of this wave
  const int koff = 2 * half;

  for (int colbase = 0; colbase < KCODES; colbase += CT) {
    const v8f zero = {0.f, 0.f, 0.f, 0.f, 0.f, 0.f, 0.f, 0.f};
    v8f acc[4] = {zero, zero, zero, zero};   // 4 independent WMMA chains
    for (int d0 = 0; d0 < D; d0 += DC) {
      __syncthreads();                       // prior consumers done (and xs on iter 0)
      // Stage emb[colbase..+255][d0..+63] into LDS (16 B128 ops per thread).
      for (int i = tid; i < CT * DC / 4; i += 256) {
        const int code = i / (DC / 4);
        const int dd   = (i % (DC / 4)) * 4;
        const float* gsrc = emb + (size_t)(colbase + code) * D + d0 + dd;
        float* ldst = es + code * ESTRIDE + dd;
#if HAVE_ASYNC_LDS
        // CDNA5 async copy: ASYNCcnt-tracked, no VGPR round trip.
        __builtin_amdgcn_global_load_async_to_lds_b128(AS1V4(gsrc), AS3V4(ldst), 0, 0);
#else
        *(float4*)ldst = *(const float4*)gsrc;
        if (colbase + CT < KCODES)           // warm next col-tile (global_prefetch_b8)
          __builtin_prefetch(emb + (size_t)(colbase + CT + code) * D + d0 + dd, 0, 0);
#endif
      }
#if HAVE_ASYNC_LDS
      wait_asynccnt0();                      // own wave's async copies landed in LDS
#endif
      __syncthreads();                       // everyone's staging visible
#pragma unroll
      for (int kk = 0; kk < DC; kk += 4) {   // 1 A frag feeds 4 WMMAs
        v2f A  = *(const v2f*)(xrow + d0 + kk + koff);
        v2f B0 = *(const v2f*)(erow + 0 * 16 * ESTRIDE + kk + koff);
        v2f B1 = *(const v2f*)(erow + 1 * 16 * ESTRIDE + kk + koff);
        v2f B2 = *(const v2f*)(erow + 2 * 16 * ESTRIDE + kk + koff);
        v2f B3 = *(const v2f*)(erow + 3 * 16 * ESTRIDE + kk + koff);
        acc[0] = __builtin_amdgcn_wmma_f32_16x16x4_f32(false, A, false, B0, (short)0, acc[0], false, false);
        acc[1] = __builtin_amdgcn_wmma_f32_16x16x4_f32(false, A, false, B1, (short)0, acc[1], false, false);
        acc[2] = __builtin_amdgcn_wmma_f32_16x16x4_f32(false, A, false, B2, (short)0, acc[2], false, false);
        acc[3] = __builtin_amdgcn_wmma_f32_16x16x4_f32(false, A, false, B3, (short)0, acc[3], false, false);
      }
    }
    // C layout: VGPR v -> row = v + 8*half (within the 16-row tile), col = l15.
    const int colb = colbase + c * 64 + l15;
#pragma unroll
    for (int t = 0; t < 4; ++t) {
      const int col = colb + t * 16;
      const float eq = esq[col];
#pragma unroll
      for (int v = 0; v < 8; ++v) {
        const float d2 = __builtin_fmaf(-2.f, acc[t][v], eq);  // x_sq is argmin-invariant
        const bool better = d2 < minval[v];                    // '<' keeps earliest index
        minval[v] = better ? d2 : minval[v];
        minidx[v] = better ? col : minidx[v];
      }
    }
  }

  // Cross-lane argmin within each 16-lane half (xor<16 stays inside the half).
#pragma unroll
  for (int v = 0; v < 8; ++v) {
#pragma unroll
    for (int off = 1; off < 16; off <<= 1) {
      const float ov = __shfl_xor(minval[v], off, 32);
      const int   oi = __shfl_xor(minidx[v], off, 32);
      const bool better = (ov < minval[v]) || (ov == minval[v] && oi < minidx[v]);
      minval[v] = better ? ov : minval[v];
      minidx[v] = better ? oi : minidx[v];
    }
  }
  if (l15 == 0) {
#pragma unroll
    for (int v = 0; v < 8; ++v) {
      const int rowloc = 16 * r + 8 * half + v;
      redv[rowloc * 4 + c] = minval[v];
      redi[rowloc * 4 + c] = minidx[v];
    }
  }
  __syncthreads();
  if (tid < MB) {                            // combine the 4 c-waves
    float bv = redv[tid * 4];
    int   bi = redi[tid * 4];
#pragma unroll
    for (int cc = 1; cc < 4; ++cc) {
      const float ov = redv[tid * 4 + cc];
      const int   oi = redi[tid * 4 + cc];
      const bool better = (ov < bv) || (ov == bv && oi < bi);
      bv = better ? ov : bv;
      bi = better ? oi : bi;
    }
    idx_out[row0 + tid] = bi;
  }
}

// ---------------------------------------------------------------------------
// Kernel 3: gather emb[idx] -> quantized output, per-row SSE, float indices.
// One wave per row (deterministic, no float atomics).
// ---------------------------------------------------------------------------
__global__ void __launch_bounds__(256) vq_gather_kernel(
    const float* __restrict__ x, const float* __restrict__ emb,
    const int* __restrict__ idx, float* __restrict__ qout,
    float* __restrict__ idxf, float* __restrict__ row_sse) {
  const int row  = blockIdx.x * 8 + (threadIdx.x >> 5);
  const int lane = threadIdx.x & 31;
  const int code = idx[row];
  const float4* e  = (const float4*)(emb + (size_t)code * D);
  const float4* xr = (const float4*)(x + (size_t)row * D);
  float4* q = (float4*)(qout + (size_t)row * D);
  float s = 0.f;
#pragma unroll
  for (int i = 0; i < D / 4 / 32; ++i) {
    const int j = lane + i * 32;
    float4 ev = e[j];
    float4 xv = xr[j];
    q[j] = ev;                                // quantized_ste forward == quantized
    const float a = xv.x - ev.x, b = xv.y - ev.y;
    const float cdl = xv.z - ev.z, d = xv.w - ev.w;
    s += a * a + b * b + cdl * cdl + d * d;
  }
#pragma unroll
  for (int off = 16; off > 0; off >>= 1) s += __shfl_xor(s, off, 32);
  if (lane == 0) { row_sse[row] = s; idxf[row] = (float)code; }
}

// ---------------------------------------------------------------------------
// Kernel 4: fixed-order reduction of row SSE -> the three scalar losses.
// ---------------------------------------------------------------------------
__global__ void vq_finalize_kernel(const float* __restrict__ row_sse,
                                   float* __restrict__ out3) {
  __shared__ float sm[256];
  float s = 0.f;
  for (int i = threadIdx.x; i < NROWS; i += 256) s += row_sse[i];
  sm[threadIdx.x] = s;
  __syncthreads();
  for (int w = 128; w > 0; w >>= 1) {
    if (threadIdx.x < w) sm[threadIdx.x] += sm[threadIdx.x + w];
    __syncthreads();
  }
  if (threadIdx.x == 0) {
    const float mse = sm[0] / (float)NUMEL;
    const float commit = 0.25f * mse;
    out3[0] = commit;          // commitment_loss
    out3[1] = mse;             // codebook_loss
    out3[2] = commit + mse;    // total
  }
}

// ---------------------------------------------------------------------------
extern "C" void kernel_launch(void* const* d_in, const int* in_sizes, int n_in,
                              void* d_out, int out_size, void* d_ws, size_t ws_size,
                              hipStream_t stream) {
  const float* x   = (const float*)d_in[0];   // [32,512,512] f32
  const float* emb = (const float*)d_in[1];   // [8192,512]  f32

  float* out       = (float*)d_out;
  float* q_out     = out;                       // NUMEL floats
  float* idx_out_f = out + (size_t)NUMEL;       // NROWS floats
  float* loss_out  = idx_out_f + NROWS;         // 3 floats

  float* esq     = (float*)d_ws;                // KCODES floats
  int*   idx_i   = (int*)(esq + KCODES);        // NROWS ints
  float* row_sse = (float*)(idx_i + NROWS);     // NROWS floats  (160 KB total ws)

  vq_esq_kernel<<<KCODES / 8, 256, 0, stream>>>(emb, esq);

  const size_t smem =
      (size_t)(MB * XSTRIDE + CT * ESTRIDE + MB * 4) * sizeof(float) +
      (size_t)(MB * 4) * sizeof(int);           // ~136.7 KB -> 2 WGs/WGP (320 KB)
  vq_argmin_kernel<<<NROWS / MB, 256, smem, stream>>>(x, emb, esq, idx_i);

  vq_gather_kernel<<<NROWS / 8, 256, 0, stream>>>(x, emb, idx_i, q_out,
                                                  idx_out_f, row_sse);
  vq_finalize_kernel<<<1, 256, 0, stream>>>(row_sse, loss_out);
}